// SNNBlock_42949672960341
// MI455X (gfx1250) — compile-verified
//
#include <hip/hip_runtime.h>

typedef float v4f __attribute__((ext_vector_type(4)));
typedef int   v4i __attribute__((ext_vector_type(4)));

#define LIF_TAU_INV 0.5f   // 1/tau, tau = 2.0
#define LIF_VTH     1.0f   // threshold (v_reset = 0)

// gfx1250 async global->LDS path (ASYNCcnt). Guarded: falls back to a plain
// nontemporal B128 load if the builtin is absent on this toolchain.
#if defined(__AMDGCN__) && __has_builtin(__builtin_amdgcn_global_load_async_to_lds_b128)
#define LIF_ASYNC_LDS 1
#else
#define LIF_ASYNC_LDS 0
#endif

__device__ __forceinline__ void lif_wait_async0() {
#if defined(__AMDGCN__)
#if __has_builtin(__builtin_amdgcn_s_wait_asynccnt)
  __builtin_amdgcn_s_wait_asynccnt(0);
#else
  asm volatile("s_wait_asynccnt 0" ::: "memory");
#endif
#endif
}

// Fully unrolled T=16 fast path. One float4 per lane: wave32 * 16B = 512B
// per wave per memory op (B128, fully coalesced). Output planes are stored
// with non-temporal hints (write-once, 512MB >> 192MB L2).
template <int T>
__global__ __launch_bounds__(256) void lif_spikes_t(const float* __restrict__ x,
                                                    float* __restrict__ out,
                                                    long long n) {
  const long long i = (((long long)blockIdx.x * 256) + threadIdx.x) * 4;
  if (i >= n) return;

  if (i + 4 <= n) {
    // Distance prefetch of the input stream (global_prefetch_b8).
    __builtin_prefetch((const void*)(x + i + (1 << 16)), 0, 0);

    v4f xv;
#if LIF_ASYNC_LDS
    __shared__ float tile[256 * 4];
    __attribute__((address_space(3))) v4i* lslot =
        (__attribute__((address_space(3))) v4i*)(&tile[threadIdx.x * 4]);
    __builtin_amdgcn_global_load_async_to_lds_b128(
        (v4i*)(x + i), lslot, 0, 0);
    lif_wait_async0();
    xv = *(__attribute__((address_space(3))) v4f*)lslot;   // ds_load_b128
#else
    xv = __builtin_nontemporal_load((const v4f*)(x + i));
#endif

    v4f v = {0.0f, 0.0f, 0.0f, 0.0f};
    float* outp = out + i;
#pragma unroll
    for (int t = 0; t < T; ++t) {
      v4f s;
#pragma unroll
      for (int c = 0; c < 4; ++c) {
        float vv = LIF_TAU_INV * (v[c] + xv[c]);      // v + (x - v)/tau, v_reset=0
        bool fire = (vv >= LIF_VTH);                  // Heaviside(v - v_th)
        s[c] = fire ? 1.0f : 0.0f;
        v[c] = fire ? 0.0f : vv;                      // hard reset
      }
      __builtin_nontemporal_store(s, (v4f*)outp);     // global_store_b128 th:NT
      outp += n;
    }
  } else {
    // Scalar tail (n not a multiple of 4).
    for (long long j = i; j < n; ++j) {
      float xs = x[j];
      float vv = 0.0f;
      for (int t = 0; t < T; ++t) {
        vv = LIF_TAU_INV * (vv + xs);
        bool fire = (vv >= LIF_VTH);
        out[(long long)t * n + j] = fire ? 1.0f : 0.0f;
        if (fire) vv = 0.0f;
      }
    }
  }
}

// Runtime-T fallback (same math, loop not compile-time unrolled).
__global__ __launch_bounds__(256) void lif_spikes_rt(const float* __restrict__ x,
                                                     float* __restrict__ out,
                                                     long long n, int T) {
  const long long i = (((long long)blockIdx.x * 256) + threadIdx.x) * 4;
  if (i >= n) return;

  if (i + 4 <= n) {
    v4f xv = __builtin_nontemporal_load((const v4f*)(x + i));
    v4f v = {0.0f, 0.0f, 0.0f, 0.0f};
    float* outp = out + i;
#pragma unroll 4
    for (int t = 0; t < T; ++t) {
      v4f s;
#pragma unroll
      for (int c = 0; c < 4; ++c) {
        float vv = LIF_TAU_INV * (v[c] + xv[c]);
        bool fire = (vv >= LIF_VTH);
        s[c] = fire ? 1.0f : 0.0f;
        v[c] = fire ? 0.0f : vv;
      }
      __builtin_nontemporal_store(s, (v4f*)outp);
      outp += n;
    }
  } else {
    for (long long j = i; j < n; ++j) {
      float xs = x[j];
      float vv = 0.0f;
      for (int t = 0; t < T; ++t) {
        vv = LIF_TAU_INV * (vv + xs);
        bool fire = (vv >= LIF_VTH);
        out[(long long)t * n + j] = fire ? 1.0f : 0.0f;
        if (fire) vv = 0.0f;
      }
    }
  }
}

extern "C" void kernel_launch(void* const* d_in, const int* in_sizes, int n_in,
                              void* d_out, int out_size, void* d_ws, size_t ws_size,
                              hipStream_t stream) {
  const float* x = (const float*)d_in[0];      // [B, L, H] fp32, flat n elements
  float* out = (float*)d_out;                  // [T, B, L, H] fp32 spikes

  const long long n = (long long)in_sizes[0];  // 8*2048*512 = 8,388,608
  const int T = (n > 0) ? (int)((long long)out_size / n) : 0;  // 16

  const long long lanes = (n + 3) / 4;
  const unsigned block = 256;
  const unsigned grid = (unsigned)((lanes + block - 1) / block);  // 8192 blocks

  if (T == 16) {
    hipLaunchKernelGGL((lif_spikes_t<16>), dim3(grid), dim3(block), 0, stream,
                       x, out, n);
  } else if (T > 0) {
    hipLaunchKernelGGL(lif_spikes_rt, dim3(grid), dim3(block), 0, stream,
                       x, out, n, T);
  }
}